// IWT_21208548507828
// MI455X (gfx1250) — compile-verified
//
#include <hip/hip_runtime.h>

typedef float v2f __attribute__((ext_vector_type(2)));
typedef float v8f __attribute__((ext_vector_type(8)));

// Haar synthesis operand A (16x4). Only rows 0,1,8,9 are used; the D-matrix
// layout (M=r -> lanes 0-15 of VGPR r, M=r+8 -> lanes 16-31) then gives:
//   D vgpr0: lanes0-15 = form0 (row 2h,  col 2w),  lanes16-31 = form2 (row 2h+1, col 2w)
//   D vgpr1: lanes0-15 = form1 (row 2h,  col 2w+1),lanes16-31 = form3 (row 2h+1, col 2w+1)
// so one b64 store of D[0:1] writes the full 2x2 output block for 16 pixels
// with all 32 lanes active.
__device__ __forceinline__ float haarA(int m, int k) {
  if (m == 0) return 0.5f;                                   // + + + +
  if (m == 1) return (k & 1) ? -0.5f : 0.5f;                 // + - + -
  if (m == 8) return (k < 2) ? 0.5f : -0.5f;                 // + + - -
  if (m == 9) return ((k == 1) || (k == 2)) ? -0.5f : 0.5f;  // + - - +
  return 0.0f;
}

__global__ __launch_bounds__(256) void iwt_haar_wmma_kernel(
    const float* __restrict__ x, float* __restrict__ out) {
  constexpr int C = 64, H = 128, W = 128;
  constexpr int PLANE = H * W;                  // 16384 elems per channel plane
  constexpr int TILES = 8 * C * H * (W / 16);   // 524288 tiles of 16 pixels

  const int lane   = (int)(threadIdx.x & 31u);
  const int laneLo = lane & 15;
  const int half   = lane >> 4;   // 0: carries K=0,1 (LL,LH); 1: carries K=2,3 (HL,HH)

  // A operand: lane holds A[laneLo][2*half] and A[laneLo][2*half+1]
  v2f a;
  a.x = haarA(laneLo, 2 * half);
  a.y = haarA(laneLo, 2 * half + 1);

  const int totalWaves = (int)((gridDim.x * blockDim.x) >> 5);
  int wave = (int)((blockIdx.x * blockDim.x + threadIdx.x) >> 5);
  wave = __builtin_amdgcn_readfirstlane(wave);  // force scalar -> uniform loop

  const v8f cz = {};  // zero accumulator

  for (int t0 = 2 * wave; t0 < TILES; t0 += 2 * totalWaves) {
    v2f bv[2];
    size_t oidx[2];
#pragma unroll
    for (int u = 0; u < 2; ++u) {
      const int t   = t0 + u;
      const int w0  = (t & 7) << 4;        // tile start along W (8 tiles/row)
      const int row = t >> 3;              // flattened (b, c, h)
      const int h   = row & (H - 1);
      const int bc  = row >> 7;            // b*64 + c
      const int b   = bc >> 6;
      const int cc  = bc & (C - 1);
      // LL plane of channel c: input channel index b*256 + c
      const size_t inBase =
          (size_t)(b * 256 + cc) * PLANE + (size_t)h * W + (size_t)(w0 + laneLo);
      // B operand: vgpr0 = LL (half=0) / HL (half=1); vgpr1 = LH / HH
      bv[u].x = x[inBase + (size_t)(half * 64) * PLANE];
      bv[u].y = x[inBase + (size_t)(128 + half * 64) * PLANE];
      // output [8,64,256,256]: lanes 0-15 write row 2h, lanes 16-31 row 2h+1
      oidx[u] = ((size_t)bc * 256 + (size_t)(2 * h + half)) * 256 +
                (size_t)(2 * (w0 + laneLo));
    }
#pragma unroll
    for (int u = 0; u < 2; ++u) {
      v8f d = __builtin_amdgcn_wmma_f32_16x16x4_f32(
          /*neg_a=*/false, a, /*neg_b=*/false, bv[u],
          /*c_mod=*/(short)0, cz, /*reuse_a=*/false, /*reuse_b=*/false);
      v2f st;
      st.x = d[0];  // col 2w   (lanes 0-15: form0 / lanes 16-31: form2)
      st.y = d[1];  // col 2w+1 (lanes 0-15: form1 / lanes 16-31: form3)
      *(v2f*)(out + oidx[u]) = st;
    }
  }
}

extern "C" void kernel_launch(void* const* d_in, const int* in_sizes, int n_in,
                              void* d_out, int out_size, void* d_ws, size_t ws_size,
                              hipStream_t stream) {
  (void)in_sizes; (void)n_in; (void)out_size; (void)d_ws; (void)ws_size;
  const float* x = (const float*)d_in[0];
  float* out = (float*)d_out;
  // 2048 blocks * 8 waves = 16384 waves; 524288 tiles -> 16 paired iterations/wave
  iwt_haar_wmma_kernel<<<dim3(2048), dim3(256), 0, stream>>>(x, out);
}